// DurandKernerMethod_16277926052172
// MI455X (gfx1250) — compile-verified
//
#include <hip/hip_runtime.h>
#include <stdint.h>

#define ORDER 32
#define NITER 100
#define TPB   128                      // threads per block (4 wave32s)
#define ROWF  (ORDER + 1)              // 33 floats per input row
#define XSTRIDE (2 * ORDER + 2)        // 66 floats per thread's LDS x-row (bank-conflict-free)

typedef uint32_t v4u __attribute__((ext_vector_type(4)));
typedef int      v8i __attribute__((ext_vector_type(8)));
typedef int      v4i __attribute__((ext_vector_type(4)));

#define ANG_STEP 0.19634954084936207f  // pi/16
#define ANG_OFF  0.04908738521234052f  // pi/64
#define EPS2     1e-28f                // (1e-14)^2

__global__ __launch_bounds__(TPB)
void DurandKerner_gfx1250_kernel(const float* __restrict__ a,
                                 float* __restrict__ out,
                                 int batch) {
  // Dual-use LDS: first TPB*ROWF floats = coefficient staging slab,
  // then reused as per-thread complex root arrays (stride 66 floats).
  __shared__ __align__(16) float s_raw[TPB * XSTRIDE];   // 33792 B

  const int tid = threadIdx.x;
  const int gid = blockIdx.x * TPB + tid;
  const long long base_elem  = (long long)blockIdx.x * (TPB * ROWF);
  const long long total_elem = (long long)batch * ROWF;

  // ---------- Stage this block's coefficient slab into LDS ----------
#if __has_builtin(__builtin_amdgcn_tensor_load_to_lds)
  if (tid < 32) {   // one wave issues the TDM descriptor
    uint64_t ga      = (uint64_t)(uintptr_t)(a + base_elem);
    uint32_t lds_off = (uint32_t)(uintptr_t)(void*)s_raw;   // low 32 bits = LDS offset
    long long remain = total_elem - base_elem;
    uint32_t tdim0 = remain > 0 ? (uint32_t)(remain > 0x7fffffffLL ? 0x7fffffffLL : remain) : 0u;

    v4u g0;
    g0[0] = 1u;                                             // count=1, user descriptor
    g0[1] = lds_off;                                        // lds_addr (bytes)
    g0[2] = (uint32_t)ga;                                   // global_addr[31:0]
    g0[3] = (uint32_t)((ga >> 32) & 0x01ffffffu) | (2u << 30); // global_addr[56:32] | type=2

    v8i g1;
    g1[0] = (int)(2u << 16);                                // wg_mask=0, data_size=2 (4B)
    g1[1] = (int)((tdim0 & 0xffffu) << 16);                 // tensor_dim0[15:0] @bit48
    g1[2] = (int)((tdim0 >> 16) | (1u << 16));              // tensor_dim0[31:16] | tensor_dim1=1
    g1[3] = (int)((uint32_t)(TPB * ROWF) << 16);            // tile_dim0 = 4224 @bits127:112
    g1[4] = 0;                                              // tile_dim1=0, tile_dim2=0
    g1[5] = (int)(uint32_t)(TPB * ROWF);                    // tensor_dim0_stride lo
    g1[6] = 0;
    g1[7] = 0;

    v4i z4 = {0, 0, 0, 0};
#if defined(__clang_major__) && (__clang_major__ >= 23)
    v8i z8 = {0, 0, 0, 0, 0, 0, 0, 0};
    __builtin_amdgcn_tensor_load_to_lds(g0, g1, z4, z4, z8, 0);
#else
    __builtin_amdgcn_tensor_load_to_lds(g0, g1, z4, z4, 0);
#endif
    __builtin_amdgcn_s_wait_tensorcnt(0);
  }
#else
  for (int i = tid; i < TPB * ROWF; i += TPB) {
    long long e = base_elem + i;
    s_raw[i] = (e < total_elem) ? a[e] : 0.0f;
  }
#endif
  __syncthreads();

  // ---------- Monic-normalize coefficients into registers ----------
  float an[ORDER];
  {
    const float* row = &s_raw[tid * ROWF];
    float a0inv = 1.0f / row[0];
#pragma unroll
    for (int k = 0; k < ORDER; ++k) an[k] = row[k + 1] * a0inv;
  }
  __syncthreads();   // staging region is about to be overwritten by x rows

  // ---------- Initial guesses on a circle ----------
  float radius = 0.0f;
#pragma unroll
  for (int j = 1; j < ORDER; ++j) {
    float e = 1.0f / (float)(j + 1);
    radius = fmaxf(radius, 2.0f * __powf(fabsf(an[j]), e));
  }
  float center = -an[0] * (1.0f / (float)ORDER);

  float2* sx = (float2*)&s_raw[tid * XSTRIDE];   // this thread's 32 complex roots
#pragma unroll 1
  for (int m = 0; m < ORDER; ++m) {
    float s, c;
    __sincosf((float)m * ANG_STEP + ANG_OFF, &s, &c);
    float2 xm;
    xm.x = center + radius * c;
    xm.y = center + radius * s;
    sx[m] = xm;
  }

  // ---------- Durand-Kerner sweeps (Gauss-Seidel order, as reference) ----------
  float md2 = 0.0f;
  bool conv = false;
#pragma unroll 1
  for (int it = 0; it < NITER; ++it) {
    md2 = 0.0f;
#pragma unroll 1
    for (int m = 0; m < ORDER; ++m) {
      float2 xm = sx[m];

      // numer = p(xm), monic Horner with real coefficients (chain A)
      float pr = 1.0f, pi = 0.0f;
#pragma unroll
      for (int k = 0; k < ORDER; ++k) {
        float nr = pr * xm.x - pi * xm.y + an[k];
        pi = pr * xm.y + pi * xm.x;
        pr = nr;
      }

      // denom = prod_j (xm - x_j + [j==m])  (chain B, independent of A)
      float dr = 1.0f, di = 0.0f;
#pragma unroll
      for (int j = 0; j < ORDER; ++j) {
        float2 xj = sx[j];
        float oh = (j == m) ? 1.0f : 0.0f;     // uniform: SALU select
        float wr = xm.x - xj.x + oh;           // xm - x[m] == 0 exactly -> w = 1
        float wi = xm.y - xj.y;
        float t = dr * wr - di * wi;
        di = dr * wi + di * wr;
        dr = t;
      }

      // delta = numer / denom; x[m] -= delta
      float den = dr * dr + di * di;
      float inv = __builtin_amdgcn_rcpf(den);
      float qr = (pr * dr + pi * di) * inv;
      float qi = (pi * dr - pr * di) * inv;
      xm.x -= qr;
      xm.y -= qi;
      sx[m] = xm;

      // each root updates once per sweep => max|y-x| == max_m |delta_m|
      md2 = fmaxf(md2, qr * qr + qi * qi);
    }
    if (md2 <= EPS2) { conv = true; break; }   // fp32 fixed point; matches reference flag
  }
  if (!conv) conv = (md2 <= EPS2);

  // ---------- Write results: complex64-interleaved roots, then flags ----------
  if (gid < batch) {
    float2* outx = (float2*)out;
    size_t obase = (size_t)gid * ORDER;
#pragma unroll 1
    for (int m = 0; m < ORDER; ++m) outx[obase + m] = sx[m];
    float* convp = out + (size_t)batch * (2 * ORDER);
    convp[gid] = conv ? 1.0f : 0.0f;
  }
}

extern "C" void kernel_launch(void* const* d_in, const int* in_sizes, int n_in,
                              void* d_out, int out_size, void* d_ws, size_t ws_size,
                              hipStream_t stream) {
  (void)n_in; (void)out_size; (void)d_ws; (void)ws_size;
  const float* a = (const float*)d_in[0];
  float* out = (float*)d_out;
  int batch = in_sizes[0] / ROWF;               // 262144
  int blocks = (batch + TPB - 1) / TPB;         // 2048
  DurandKerner_gfx1250_kernel<<<blocks, TPB, 0, stream>>>(a, out, batch);
}